// MultiHeadAttentionQuantum_65481071396889
// MI455X (gfx1250) — compile-verified
//
#include <hip/hip_runtime.h>
#include <hip/hip_bf16.h>
#include <math.h>

// MI455X / gfx1250, wave32. Sizes fixed by the reference:
// B=4, S=4096, EMBED=512, HEADS=4, NQ=4  -> tokens N = 16384, d_k = 1.
#define NTOK   16384
#define EMBED  512
#define SEQ    4096

typedef float v2f __attribute__((ext_vector_type(2)));
typedef float v8f __attribute__((ext_vector_type(8)));
typedef unsigned int u32x4 __attribute__((ext_vector_type(4)));
typedef int          i32x4 __attribute__((ext_vector_type(4)));
typedef int          i32x8 __attribute__((ext_vector_type(8)));

#if __has_builtin(__builtin_amdgcn_tensor_load_to_lds) && \
    __has_builtin(__builtin_amdgcn_s_wait_tensorcnt)
#define HAVE_TDM 1
#else
#define HAVE_TDM 0
#endif

// TDM tile geometry for kernel B: 16 rows x 512 f32 per wave, LDS rows padded
// by 2 DWORDs every 256 DWORDs (D# pad feature) -> row stride 516 DWORDs.
// 516 mod 64 = 4 banks/row: the 16-lane column read of the WMMA A-fragment
// hits 16 distinct banks (4*m), and the b64 pairs tile all 64 banks.
#define ROW_DW   516
#define ROW_B    (ROW_DW * 4)          // 2064 bytes
#define WTILE_B  (16 * ROW_B)          // 33024 bytes per wave
#define BWAVES   4                     // waves per block in kernel B
#define BLDS_B   (BWAVES * WTILE_B)    // 132096 bytes dynamic LDS

// ---------------------------------------------------------------------------
// Kernel A: fuse Wqp=Wq@Wp, Wkp=Wk@Wp, Wvp=Wv@Wp (each 512x4) and store them
// directly in V_WMMA_F32_16X16X4_F32 B-fragment lane order:
//   Bfrag[(K*32+lane)*2+half] = Bpacked[4K + hi*2 + half][n]
//   n=lane&15 (col: 0-3 q, 4-7 k, 8-11 v, 12-15 zero), hi=lane>>4.
// ---------------------------------------------------------------------------
__global__ void fuse_weights_kernel(const float* __restrict__ Wq,
                                    const float* __restrict__ Wk,
                                    const float* __restrict__ Wv,
                                    const float* __restrict__ Wp,
                                    float* __restrict__ Bfrag) {
    int pos  = blockIdx.x * blockDim.x + threadIdx.x;   // 0..8191
    int half = pos & 1;
    int lane = (pos >> 1) & 31;
    int K    = pos >> 6;                                // 0..127
    int n    = lane & 15;
    int hi   = lane >> 4;
    int e    = K * 4 + hi * 2 + half;                   // 0..511
    float acc = 0.0f;
    if (n < 12) {
        const float* W = (n < 4) ? Wq : (n < 8) ? Wk : Wv;
        const float* wr = W + e * EMBED;
        int i = n & 3;
        #pragma unroll 8
        for (int j = 0; j < EMBED; ++j)
            acc = fmaf(wr[j], Wp[j * 4 + i], acc);
    }
    Bfrag[pos] = acc;
}

// ---------------------------------------------------------------------------
// Kernel B: q/k/v projection via V_WMMA_F32_16X16X4_F32 (K=512, 128 steps)
// fused with the closed-form quantum feature map  <Z_i> = cos(proj_i + th_i).
// One wave owns 16 tokens. A-tile staged LDS-side by the Tensor Data Mover:
// each wave issues one tensor_load_to_lds (2D D#: tile 512x16 f32, tensor
// stride 512, LDS pad 2 DW / 256 DW), waits s_wait_tensorcnt 0, then streams
// conflict-free ds_load_b64 A-fragments. B-fragments (shared by all waves)
// stream from L0/L2 as coalesced b64.
// ---------------------------------------------------------------------------
__global__ void qkv_quantum_kernel(const float* __restrict__ x,
                                   const float* __restrict__ Bfrag,
                                   const float* __restrict__ theta,
                                   float* __restrict__ qqm,
                                   float* __restrict__ kqm,
                                   float* __restrict__ vqm) {
    int lane = threadIdx.x & 31;
    int winb = threadIdx.x >> 5;                        // wave in block
    int wave = blockIdx.x * BWAVES + winb;
    int tok0 = wave * 16;
    int m  = lane & 15;
    int hi = lane >> 4;

    v8f c = {0.f, 0.f, 0.f, 0.f, 0.f, 0.f, 0.f, 0.f};

#if HAVE_TDM
    extern __shared__ float smem[];
    unsigned lds_off = (unsigned)(size_t)(void*)smem + (unsigned)(winb * WTILE_B);
    unsigned long long ga =
        (unsigned long long)(size_t)(const void*)x + (unsigned long long)tok0 * (EMBED * 4ull);

    // ---- D# group 0: count=1 | lds_addr | global_addr | type=2 ----
    u32x4 g0;
    g0[0] = 1u;                                         // count=1, user mode
    g0[1] = lds_off;                                    // lds_addr (bytes)
    g0[2] = (unsigned)(ga & 0xFFFFFFFFull);             // global_addr[31:0]
    g0[3] = (unsigned)((ga >> 32) & 0x1FFFFFFull)       // global_addr[56:32]
          | (2u << 30);                                 // type=2 (image)
    // ---- D# group 1: sizes/strides/pad ----
    i32x8 g1;
    g1[0] = (2 << 16)        // data_size = 4 bytes
          | (1 << 20)        // pad_enable
          | (7 << 22)        // pad_interval: 256 DWORDs
          | (1 << 25);       // pad_amount:   2 DWORDs
    g1[1] = (EMBED & 0xFFFF) << 16;     // tensor_dim0[15:0] = 512  (bits 63:48)
    g1[2] = (16 << 16);                 // dim0 hi=0 | tensor_dim1[15:0]=16 (95:80)
    g1[3] = (EMBED << 16);              // dim1 hi=0 | tile_dim0=512 (127:112)
    g1[4] = 16;                         // tile_dim1=16 (143:128), tile_dim2=0
    g1[5] = EMBED;                      // tensor_dim0_stride low32 = 512
    g1[6] = 0;                          // stride0 hi | stride1 lo
    g1[7] = 0;                          // stride1 hi
    i32x4 gz4 = {0, 0, 0, 0};           // groups 2/3 unused (2D tensor)
    i32x8 gz8 = {0, 0, 0, 0, 0, 0, 0, 0};

    __builtin_amdgcn_tensor_load_to_lds(g0, g1, gz4, gz4, gz8, 0);
    __builtin_amdgcn_s_wait_tensorcnt((short)0);

    const float* arow = smem + winb * (WTILE_B / 4) + m * ROW_DW + hi * 2;
    #pragma unroll 4
    for (int k = 0; k < EMBED; k += 4) {
        int koff = k + 2 * (k >> 8);                    // skip LDS pad DWORDs
        v2f a = *(const v2f*)(arow + koff);
        v2f b = *(const v2f*)(Bfrag + (((k >> 2) * 32 + lane) << 1));
        c = __builtin_amdgcn_wmma_f32_16x16x4_f32(
                false, a, false, b, (short)0, c, false, false);
    }
#else
    const float* xrow = x + (size_t)(tok0 + m) * EMBED + hi * 2;
    #pragma unroll 4
    for (int k = 0; k < EMBED; k += 4) {
        v2f a = *(const v2f*)(xrow + k);
        v2f b = *(const v2f*)(Bfrag + (((k >> 2) * 32 + lane) << 1));
        c = __builtin_amdgcn_wmma_f32_16x16x4_f32(
                false, a, false, b, (short)0, c, false, false);
    }
#endif

    // C layout: VGPR j -> row M=j (lanes 0-15) / M=j+8 (lanes 16-31), col N=m.
    if (m < 12) {
        float* dst = (m < 4) ? qqm : (m < 8) ? kqm : vqm;
        int   i  = m & 3;
        float th = theta[i];
        #pragma unroll
        for (int j = 0; j < 8; ++j) {
            int tok = tok0 + (hi ? j + 8 : j);
            dst[tok * 4 + i] = __cosf(c[j] + th);  // native v_cos path; |arg| ~ few
        }
    }
}

// ---------------------------------------------------------------------------
// Kernel C1: per (b,h) key moments  N_m = sum_k k^m,  M_m = sum_k k^m * v,
// m = 0..7 (|score| <= 1/sqrt(128) -> 8-term Taylor exact to f32).
// ---------------------------------------------------------------------------
__global__ void moments_kernel(const float* __restrict__ kqm,
                               const float* __restrict__ vqm,
                               float* __restrict__ mom) {
    int bh = blockIdx.x;           // 0..15
    int b  = bh >> 2, h = bh & 3;
    int tid = threadIdx.x;

    float acc[16];
    #pragma unroll
    for (int i = 0; i < 16; ++i) acc[i] = 0.0f;

    for (int s = tid; s < SEQ; s += 256) {
        int base = (b * SEQ + s) * 4 + h;
        float kv = kqm[base];
        float vv = vqm[base];
        float p = 1.0f;
        #pragma unroll
        for (int m2 = 0; m2 < 8; ++m2) {
            acc[m2]     += p;                            // N_m
            acc[8 + m2] = fmaf(p, vv, acc[8 + m2]);      // M_m
            p *= kv;
        }
    }

    __shared__ float red[256];
    #pragma unroll
    for (int v = 0; v < 16; ++v) {
        red[tid] = acc[v];
        __syncthreads();
        for (int s2 = 128; s2 > 0; s2 >>= 1) {
            if (tid < s2) red[tid] += red[tid + s2];
            __syncthreads();
        }
        if (tid == 0) mom[bh * 16 + v] = red[0];
        __syncthreads();
    }
}

// ---------------------------------------------------------------------------
// Kernel C2: attn[token,h] = (sum_m t^m/m! * M_m) / (sum_m t^m/m! * N_m),
// t = q_val / sqrt(128). Exact softmax-average to f32 precision.
// ---------------------------------------------------------------------------
__global__ void attn_kernel(const float* __restrict__ qqm,
                            const float* __restrict__ mom,
                            float* __restrict__ attn) {
    int idx = blockIdx.x * blockDim.x + threadIdx.x;  // token*4 + h
    int h     = idx & 3;
    int token = idx >> 2;
    int b     = token >> 12;
    int bh    = b * 4 + h;

    const float c0 = 0.0883883476483184f;   // 1/sqrt(128)
    const float inv_fact[8] = {1.f, 1.f, 0.5f, 1.f/6.f, 1.f/24.f,
                               1.f/120.f, 1.f/720.f, 1.f/5040.f};
    float t = c0 * qqm[idx];
    float num = 0.f, den = 0.f, p = 1.f;
    #pragma unroll
    for (int m2 = 0; m2 < 8; ++m2) {
        float w = p * inv_fact[m2];
        den = fmaf(w, mom[bh * 16 + m2],     den);
        num = fmaf(w, mom[bh * 16 + 8 + m2], num);
        p *= t;
    }
    attn[idx] = num / den;
}

// ---------------------------------------------------------------------------
// Kernel D: out = attn(16384x4) @ Wo(4x512) via V_WMMA_F32_16X16X4_F32.
// K = 4 exactly -> one WMMA per 16x16 output tile.
// ---------------------------------------------------------------------------
__global__ void out_gemm_kernel(const float* __restrict__ attn,
                                const float* __restrict__ Wo,
                                float* __restrict__ out) {
    int lane = threadIdx.x & 31;
    int w    = blockIdx.x * 8 + (threadIdx.x >> 5);
    int ttile = w >> 5;          // 0..1023 token tile
    int ctile = w & 31;          // 0..31  column tile
    int m  = lane & 15;
    int hi = lane >> 4;

    int tok = ttile * 16 + m;
    v2f a = *(const v2f*)(attn + tok * 4 + (hi ? 2 : 0));   // A 16x4

    int col = ctile * 16 + m;
    v2f b;
    b.x = Wo[(hi * 2)     * EMBED + col];   // B rows K={0,2}
    b.y = Wo[(hi * 2 + 1) * EMBED + col];   // B rows K={1,3}

    v8f c = {0.f, 0.f, 0.f, 0.f, 0.f, 0.f, 0.f, 0.f};
    c = __builtin_amdgcn_wmma_f32_16x16x4_f32(
            false, a, false, b, (short)0, c, false, false);

    #pragma unroll
    for (int j = 0; j < 8; ++j) {
        int row = ttile * 16 + (hi ? j + 8 : j);
        out[(size_t)row * EMBED + col] = c[j];
    }
}

// ---------------------------------------------------------------------------
// Launch
// ---------------------------------------------------------------------------
extern "C" void kernel_launch(void* const* d_in, const int* in_sizes, int n_in,
                              void* d_out, int out_size, void* d_ws, size_t ws_size,
                              hipStream_t stream) {
    const float* x     = (const float*)d_in[0];
    const float* Wq    = (const float*)d_in[1];
    const float* Wk    = (const float*)d_in[2];
    const float* Wv    = (const float*)d_in[3];
    const float* Wp    = (const float*)d_in[4];
    const float* Wo    = (const float*)d_in[5];
    const float* theta = (const float*)d_in[6];
    float* out = (float*)d_out;

    float* ws    = (float*)d_ws;
    float* Bfrag = ws;                     //  8192 floats
    float* qqm   = Bfrag + 8192;           // 65536
    float* kqm   = qqm   + NTOK * 4;       // 65536
    float* vqm   = kqm   + NTOK * 4;       // 65536
    float* mom   = vqm   + NTOK * 4;       //   256
    float* attn  = mom   + 256;            // 65536

    // A: fused projection weights into WMMA B-fragment order
    fuse_weights_kernel<<<32, 256, 0, stream>>>(Wq, Wk, Wv, Wp, Bfrag);

    // B: TDM-staged WMMA projection (K=512) + quantum cos. 1024 waves,
    //    4 waves/block, 132KB dynamic LDS (2 blocks/WGP within 320KB).
    qkv_quantum_kernel<<<256, 32 * BWAVES, HAVE_TDM ? BLDS_B : 0, stream>>>(
        x, Bfrag, theta, qqm, kqm, vqm);

    // C1: key moments per (b,h)
    moments_kernel<<<16, 256, 0, stream>>>(kqm, vqm, mom);

    // C2: Taylor softmax-average (exact to f32 for |score| <= 1/sqrt(128))
    attn_kernel<<<256, 256, 0, stream>>>(qqm, mom, attn);

    // D: output GEMM via WMMA, K=4 exact
    out_gemm_kernel<<<4096, 256, 0, stream>>>(attn, Wo, out);
}